// ChamferDeviationL2_85143431676065
// MI455X (gfx1250) — compile-verified
//
#include <hip/hip_runtime.h>

typedef __attribute__((ext_vector_type(2))) float v2f;
typedef __attribute__((ext_vector_type(8))) float v8f;

#define CHUNK 2048            // target points staged in LDS per pass (32 KB as float4)
#define TILE 16
#define WAVES 8
#define ROWS_PER_BLOCK (WAVES * TILE)   // 128 query rows per workgroup

__global__ void chamfer_init(float* acc) {
    if (threadIdx.x < 8) acc[threadIdx.x] = 0.0f;
}

// One direction of chamfer: for each point of P find nearest point of Q,
// accumulate sum(min_dist) and sum(p - q_nn) into acc[0..3] (global atomics).
__global__ __launch_bounds__(256)
void chamfer_dir(const float* __restrict__ P, const float* __restrict__ Q,
                 int Npts, int Mpts, float* __restrict__ acc) {
    __shared__ float4 qbuf[CHUNK];   // (x, y, z, |q|^2)
    __shared__ float  lacc[4];

    const int lane  = threadIdx.x & 31;
    const int wave  = threadIdx.x >> 5;
    const int batch = blockIdx.y;
    const int rowBase = blockIdx.x * ROWS_PER_BLOCK + wave * TILE;
    const bool hi  = lane >= 16;
    const int  l15 = lane & 15;

    if (threadIdx.x < 4) lacc[threadIdx.x] = 0.0f;

    const float* Pb = P + (size_t)batch * (size_t)Npts * 3;
    const float* Qb = Q + (size_t)batch * (size_t)Mpts * 3;

    // ---- A-matrix operand: this wave's 16 query points, 16x4 f32 layout ----
    // lanes 0-15: VGPR0=K0(x), VGPR1=K1(y); lanes 16-31: VGPR0=K2(z), VGPR1=K3(0)
    const int m0 = rowBase + l15;
    const float px = Pb[m0 * 3 + 0];
    const float py = Pb[m0 * 3 + 1];
    const float pz = Pb[m0 * 3 + 2];
    v2f a;
    a.x = hi ? pz : px;
    a.y = hi ? 0.0f : py;
    const float sqOwn = px * px + py * py + pz * pz;

    // sqP[v] = |P[rowBase + v + (hi?8:0)]|^2, matching C/D row layout
    float sqP[8];
#pragma unroll
    for (int v = 0; v < 8; ++v)
        sqP[v] = __shfl(sqOwn, v + (hi ? 8 : 0), 32);

    float dmin[8];
    int   imin[8];
#pragma unroll
    for (int v = 0; v < 8; ++v) { dmin[v] = 3.4e38f; imin[v] = 0; }

    for (int chunk = 0; chunk < Mpts; chunk += CHUNK) {
        __syncthreads();
        // Cooperative stage of target points into LDS with precomputed |q|^2
        for (int j = threadIdx.x; j < CHUNK; j += blockDim.x) {
            const int n = chunk + j;
            const float qx = Qb[n * 3 + 0];
            const float qy = Qb[n * 3 + 1];
            const float qz = Qb[n * 3 + 2];
            qbuf[j] = make_float4(qx, qy, qz, qx * qx + qy * qy + qz * qz);
        }
        __syncthreads();

        // 128 tiles per chunk; unroll 4 so independent WMMAs (fresh C each
        // tile) issue back-to-back and their D->VALU hazard is hidden by the
        // neighbouring tiles' min/argmin VALU work.
#pragma unroll 4
        for (int t = 0; t < CHUNK; t += TILE) {
            // ---- B-matrix operand: 16 target points, 4x16 f32 layout ----
            const float4 q = qbuf[t + l15];
            v2f b;
            b.x = hi ? q.z : q.x;    // K0 row = x (low lanes) / K2 row = z (high lanes)
            b.y = hi ? 0.0f : q.y;   // K1 row = y / K3 row = 0
            v8f c = {};
            // inner[M][N] = x*x + y*y + z*z  (K=3 zero-padded to 4)
            c = __builtin_amdgcn_wmma_f32_16x16x4_f32(
                    false, a, false, b, (short)0, c, false, false);

            const int n = chunk + t + l15;
#pragma unroll
            for (int v = 0; v < 8; ++v) {
                const float d = fmaf(c[v], -2.0f, sqP[v] + q.w);
                if (d < dmin[v]) { dmin[v] = d; imin[v] = n; }
            }
        }
    }

    // Butterfly (min, argmin) reduction across each 16-lane half (wave32).
#pragma unroll
    for (int off = 8; off >= 1; off >>= 1) {
#pragma unroll
        for (int v = 0; v < 8; ++v) {
            const float d2 = __shfl_xor(dmin[v], off, 32);
            const int   i2 = __shfl_xor(imin[v], off, 32);
            if (d2 < dmin[v] || (d2 == dmin[v] && i2 < imin[v])) {
                dmin[v] = d2; imin[v] = i2;
            }
        }
    }

    // Lane 0 owns rows rowBase+0..7, lane 16 owns rows rowBase+8..15.
    if (lane == 0 || lane == 16) {
        float sd = 0.0f, sx = 0.0f, sy = 0.0f, sz = 0.0f;
#pragma unroll
        for (int v = 0; v < 8; ++v) {
            const int mm = rowBase + v + (hi ? 8 : 0);
            const float mx = Pb[mm * 3 + 0];
            const float my = Pb[mm * 3 + 1];
            const float mz = Pb[mm * 3 + 2];
            const int nn = imin[v];
            sd += dmin[v];
            sx += mx - Qb[nn * 3 + 0];
            sy += my - Qb[nn * 3 + 1];
            sz += mz - Qb[nn * 3 + 2];
        }
        atomicAdd(&lacc[0], sd);
        atomicAdd(&lacc[1], sx);
        atomicAdd(&lacc[2], sy);
        atomicAdd(&lacc[3], sz);
    }
    __syncthreads();
    if (threadIdx.x < 4) atomicAdd(&acc[threadIdx.x], lacc[threadIdx.x]);
}

__global__ void chamfer_final(const float* __restrict__ acc, float* __restrict__ out,
                              float invBN, float invBM) {
    if (threadIdx.x == 0) {
        out[0] = acc[0] * invBN + acc[4] * invBM;   // mean(dist1)+mean(dist2)
        out[1] = acc[1] * invBN + acc[5] * invBM;   // mean(dev1.x)+mean(dev2.x)
        out[2] = acc[2] * invBN + acc[6] * invBM;
        out[3] = acc[3] * invBN + acc[7] * invBM;
    }
}

extern "C" void kernel_launch(void* const* d_in, const int* in_sizes, int n_in,
                              void* d_out, int out_size, void* d_ws, size_t ws_size,
                              hipStream_t stream) {
    const float* xyz1 = (const float*)d_in[0];
    const float* xyz2 = (const float*)d_in[1];
    const int B = 8;
    const int N = in_sizes[0] / (B * 3);
    const int M = in_sizes[1] / (B * 3);

    float* acc = (float*)d_ws;   // 8 floats: [sum_d1, sum_dev1.xyz, sum_d2, sum_dev2.xyz]

    chamfer_init<<<1, 32, 0, stream>>>(acc);
    chamfer_dir<<<dim3(N / ROWS_PER_BLOCK, B), 256, 0, stream>>>(xyz1, xyz2, N, M, acc);
    chamfer_dir<<<dim3(M / ROWS_PER_BLOCK, B), 256, 0, stream>>>(xyz2, xyz1, M, N, acc + 4);
    chamfer_final<<<1, 32, 0, stream>>>(acc, (float*)d_out,
                                        1.0f / (float)(B * N), 1.0f / (float)(B * M));
}